// BinaryMasking_64665027608684
// MI455X (gfx1250) — compile-verified
//
#include <hip/hip_runtime.h>
#include <stdint.h>

// Problem constants (match reference)
#define T_DIM 32
#define H_DIM 64
#define W_DIM 64
#define NUM_TOKENS (T_DIM * H_DIM * W_DIM)   // 131072
#define B_DIM 64
#define EPS_F 1e-3f
#define CAND_CAP 4096
#define TILE 4096                            // tokens staged per block in pass 1

// ---------- helpers ----------

__device__ __forceinline__ float clampu(float u) {
    return fminf(fmaxf(u, EPS_F), 1.0f - EPS_F);
}

// Monotone map float -> uint32 so that unsigned compare == float compare.
__device__ __forceinline__ unsigned int order_key(float f) {
    unsigned int u = __float_as_uint(f);
    return (u & 0x80000000u) ? ~u : (u | 0x80000000u);
}

// LDS byte offset of a __shared__ object: low 32 bits of the generic pointer
// are the AS3 offset (flat LDS aperture discards the upper bits).
__device__ __forceinline__ unsigned int lds_byte_off(const void* p) {
    return (unsigned int)(uintptr_t)p;
}

__device__ __forceinline__ float score_term(float tok, float cd, float lth) {
    return logf(clampu(tok)) + lth + logf(fmaxf(cd, 1e-12f));
}

// ---------- pass 1: compute keys (b128 stores), build 16-bit-MSB histogram ----------
// grid = B_DIM * (NUM_TOKENS/TILE) blocks of 256 threads.
__global__ void bm_pass1_keys_hist(const float* __restrict__ U_token,
                                   const float* __restrict__ U_t,
                                   const float* __restrict__ U_h,
                                   const float* __restrict__ cond,
                                   unsigned int* __restrict__ keys,
                                   unsigned int* __restrict__ hist) {
    __shared__ float4 s_tok[TILE / 4];    // 16 KB
    __shared__ float4 s_cond[TILE / 4];   // 16 KB
    __shared__ float  s_lt[T_DIM];
    __shared__ float  s_lh[H_DIM];

    const int tid  = threadIdx.x;
    const int tile = blockIdx.x & 31;   // 32 tiles per row
    const int row  = blockIdx.x >> 5;
    const int base = tile * TILE;

    const float* gTok  = U_token + (size_t)row * NUM_TOKENS + base;
    const float* gCond = cond    + (size_t)row * NUM_TOKENS + base;

    // per-dim prior terms (tiny, plain loads)
    if (tid < T_DIM) {
        s_lt[tid] = 0.5f * logf(clampu(U_t[row * T_DIM + tid]));
    } else if (tid < T_DIM + H_DIM) {
        int h = tid - T_DIM;
        s_lh[h] = 0.25f * logf(clampu(U_h[row * H_DIM + h]));
    }

    // Async stage the two 16 KB tiles into LDS (CDNA5 ASYNCcnt path).
    // Each lane loads 16 B; consecutive lanes 16 B apart -> 512 B / wave instr.
    unsigned int lt_ = lds_byte_off(&s_tok[0]);
    unsigned int lc_ = lds_byte_off(&s_cond[0]);
#pragma unroll
    for (int k = 0; k < 4; ++k) {
        int g = k * 256 + tid;           // float4 group index [0, 1024)
        unsigned long long at = (unsigned long long)(uintptr_t)(gTok  + g * 4);
        unsigned long long ac = (unsigned long long)(uintptr_t)(gCond + g * 4);
        unsigned int dt = lt_ + (unsigned int)(g * 16);
        unsigned int dc = lc_ + (unsigned int)(g * 16);
        asm volatile("global_load_async_to_lds_b128 %0, %1, off"
                     :: "v"(dt), "v"(at) : "memory");
        asm volatile("global_load_async_to_lds_b128 %0, %1, off"
                     :: "v"(dc), "v"(ac) : "memory");
    }
    asm volatile("s_wait_asynccnt 0" ::: "memory");
    __syncthreads();

    uint4* keys4 = (uint4*)keys;
    unsigned int* hrow = hist + ((size_t)row << 16);

    // Groups of 4 consecutive tokens never straddle a t- or h-boundary
    // (both periods are multiples of 4), so one prior lookup per group.
#pragma unroll
    for (int j = 0; j < 4; ++j) {
        int g  = j * 256 + tid;          // group index within tile
        int n0 = base + g * 4;           // first token of group
        int t  = n0 >> 12;               // n / (H*W)
        int h  = (n0 >> 6) & (H_DIM - 1);
        float lth = s_lt[t] + s_lh[h];
        float4 tk = s_tok[g];
        float4 cd = s_cond[g];
        uint4 kv;
        kv.x = order_key(score_term(tk.x, cd.x, lth));
        kv.y = order_key(score_term(tk.y, cd.y, lth));
        kv.z = order_key(score_term(tk.z, cd.z, lth));
        kv.w = order_key(score_term(tk.w, cd.w, lth));
        keys4[(size_t)row * (NUM_TOKENS / 4) + tile * (TILE / 4) + g] = kv;
        atomicAdd(&hrow[kv.x >> 16], 1u);
        atomicAdd(&hrow[kv.y >> 16], 1u);
        atomicAdd(&hrow[kv.z >> 16], 1u);
        atomicAdd(&hrow[kv.w >> 16], 1u);
    }
}

// ---------- pass 2: per-row descending histogram select -> (bin, residual rank) ----------
// Coalesced: chunk c loads h[c*256 + tid] (contiguous across lanes), wave32
// shfl reduction -> per-wave partials -> segment sums.
__global__ void bm_pass2_find_bin(const unsigned int* __restrict__ hist,
                                  const float* __restrict__ U_rate,
                                  int2* __restrict__ meta) {
    __shared__ unsigned int s_wp[256 * 8];   // 8 KB: per-chunk per-wave partials
    __shared__ unsigned int s_part[256];
    __shared__ unsigned int s_pref[256];
    const int row  = blockIdx.x;
    const int tid  = threadIdx.x;
    const int lane = tid & 31;
    const int wave = tid >> 5;

    // K for this row (stratified rate math, float32 then trunc toward zero)
    float u = U_rate[0];
    float L = (float)row / (float)(B_DIM - 1);
    float m = L + u;
    m -= floorf(m);
    m = fminf(fmaxf(m, EPS_F), 1.0f - EPS_F);
    float R = m * 0.8f + 0.1f;
    int K = (int)(R * (float)NUM_TOKENS);

    const unsigned int* h = hist + ((size_t)row << 16);

    // segment s = bins [s*256, s*256+256); chunk c loads segment c coalesced
    for (int c = 0; c < 256; ++c) {
        unsigned int v = h[c * 256 + tid];
#pragma unroll
        for (int off = 16; off > 0; off >>= 1) v += __shfl_down(v, off, 32);
        if (lane == 0) s_wp[c * 8 + wave] = v;
    }
    __syncthreads();
    unsigned int part = 0;
#pragma unroll
    for (int w = 0; w < 8; ++w) part += s_wp[tid * 8 + w];
    s_part[tid] = part;
    __syncthreads();
    if (tid == 0) {
        unsigned int acc = 0;
        for (int s = 255; s >= 0; --s) { s_pref[s] = acc; acc += s_part[s]; }
    }
    __syncthreads();

    unsigned int pref = s_pref[tid];   // count in bins >= (tid+1)*256
    if ((int)pref < K && K <= (int)(pref + part)) {
        unsigned int cum = pref;
        for (int l = 255; l >= 0; --l) {
            int bin = tid * 256 + l;
            unsigned int c = h[bin];
            if ((int)(cum + c) >= K) {
                meta[row] = make_int2(bin, K - (int)cum);  // residual rank >= 1
                break;
            }
            cum += c;
        }
    }
}

// ---------- pass 3: gather candidates sharing the threshold 16-bit prefix ----------
__global__ void bm_pass3_candidates(const unsigned int* __restrict__ keys,
                                    const int2* __restrict__ meta,
                                    unsigned int* __restrict__ cand,
                                    unsigned int* __restrict__ candCount) {
    const uint4* keys4 = (const uint4*)keys;
    size_t stride = (size_t)gridDim.x * blockDim.x;
    size_t total4 = (size_t)B_DIM * (NUM_TOKENS / 4);
    for (size_t i = (size_t)blockIdx.x * blockDim.x + threadIdx.x; i < total4; i += stride) {
        int row = (int)(i >> 15);               // / (NUM_TOKENS/4)
        int p = meta[row].x;
        uint4 kv = keys4[i];
        unsigned int* cc = &candCount[row];
        unsigned int* cr = cand + (size_t)row * CAND_CAP;
        if ((int)(kv.x >> 16) == p) { unsigned int q = atomicAdd(cc, 1u); if (q < CAND_CAP) cr[q] = kv.x; }
        if ((int)(kv.y >> 16) == p) { unsigned int q = atomicAdd(cc, 1u); if (q < CAND_CAP) cr[q] = kv.y; }
        if ((int)(kv.z >> 16) == p) { unsigned int q = atomicAdd(cc, 1u); if (q < CAND_CAP) cr[q] = kv.z; }
        if ((int)(kv.w >> 16) == p) { unsigned int q = atomicAdd(cc, 1u); if (q < CAND_CAP) cr[q] = kv.w; }
    }
}

// ---------- pass 4: r-th largest among candidates (LDS-staged) -> threshold key ----------
__global__ void bm_pass4_threshold(const unsigned int* __restrict__ cand,
                                   const unsigned int* __restrict__ candCount,
                                   const int2* __restrict__ meta,
                                   unsigned int* __restrict__ thresh) {
    __shared__ unsigned int s_cand[CAND_CAP];   // 16 KB
    const int row = blockIdx.x;
    unsigned int ccv = candCount[row];
    int n = (int)(ccv < (unsigned int)CAND_CAP ? ccv : (unsigned int)CAND_CAP);
    int r = meta[row].y;
    if (r > n) r = n;
    for (int i = threadIdx.x; i < n; i += blockDim.x)
        s_cand[i] = cand[(size_t)row * CAND_CAP + i];
    __syncthreads();
    if (n == 0) return;  // thresh stays 0 from init (select-all fallback)
    for (int i = threadIdx.x; i < n; i += blockDim.x) {
        unsigned int ci = s_cand[i];
        int grt = 0;
        for (int j = 0; j < n; ++j) {
            unsigned int cj = s_cand[j];
            grt += (int)((cj > ci) || (cj == ci && j < i));
        }
        if (grt == r - 1) thresh[row] = ci;  // value is order-independent
    }
}

// ---------- pass 5: write mask (b128 in / b128 out) ----------
__global__ void bm_pass5_mask(const unsigned int* __restrict__ keys,
                              const unsigned int* __restrict__ thresh,
                              float* __restrict__ out) {
    const uint4* keys4 = (const uint4*)keys;
    float4* out4 = (float4*)out;
    size_t stride = (size_t)gridDim.x * blockDim.x;
    size_t total4 = (size_t)B_DIM * (NUM_TOKENS / 4);
    for (size_t i = (size_t)blockIdx.x * blockDim.x + threadIdx.x; i < total4; i += stride) {
        unsigned int th = thresh[i >> 15];
        uint4 kv = keys4[i];
        float4 o;
        o.x = (kv.x >= th) ? 1.0f : 0.0f;
        o.y = (kv.y >= th) ? 1.0f : 0.0f;
        o.z = (kv.z >= th) ? 1.0f : 0.0f;
        o.w = (kv.w >= th) ? 1.0f : 0.0f;
        out4[i] = o;
    }
}

// ---------- init: zero scratch counters each call (graph-replay safe) ----------
__global__ void bm_init_ws(unsigned int* __restrict__ hist,
                           unsigned int* __restrict__ candCount,
                           unsigned int* __restrict__ thresh) {
    uint4* h4 = (uint4*)hist;
    size_t stride = (size_t)gridDim.x * blockDim.x;
    size_t nh4 = ((size_t)B_DIM << 16) / 4;
    uint4 z = make_uint4(0u, 0u, 0u, 0u);
    for (size_t j = (size_t)blockIdx.x * blockDim.x + threadIdx.x; j < nh4; j += stride) h4[j] = z;
    size_t i = (size_t)blockIdx.x * blockDim.x + threadIdx.x;
    if (i < B_DIM) { candCount[i] = 0u; thresh[i] = 0u; }
}

extern "C" void kernel_launch(void* const* d_in, const int* in_sizes, int n_in,
                              void* d_out, int out_size, void* d_ws, size_t ws_size,
                              hipStream_t stream) {
    (void)in_sizes; (void)n_in; (void)out_size; (void)ws_size;
    const float* U_token = (const float*)d_in[0];
    const float* U_t     = (const float*)d_in[1];
    const float* U_h     = (const float*)d_in[2];
    const float* U_rate  = (const float*)d_in[3];
    const float* cond    = (const float*)d_in[4];
    float* out = (float*)d_out;

    // Workspace layout (bytes):
    //   keys      : 0         .. 33,554,432   (B * N * 4)
    //   hist      : 33,554,432.. +16,777,216  (B * 65536 * 4)
    //   cand      : 50,331,648.. +1,048,576   (B * CAND_CAP * 4)
    //   candCount : 51,380,224.. +256
    //   meta      : 51,380,480.. +512
    //   thresh    : 51,380,992.. +256
    char* ws = (char*)d_ws;
    unsigned int* keys      = (unsigned int*)(ws);
    unsigned int* hist      = (unsigned int*)(ws + 33554432u);
    unsigned int* cand      = (unsigned int*)(ws + 50331648u);
    unsigned int* candCount = (unsigned int*)(ws + 51380224u);
    int2*         meta      = (int2*)       (ws + 51380480u);
    unsigned int* thresh    = (unsigned int*)(ws + 51380992u);

    bm_init_ws         <<<1024, 256, 0, stream>>>(hist, candCount, thresh);
    bm_pass1_keys_hist <<<B_DIM * (NUM_TOKENS / TILE), 256, 0, stream>>>(U_token, U_t, U_h, cond, keys, hist);
    bm_pass2_find_bin  <<<B_DIM, 256, 0, stream>>>(hist, U_rate, meta);
    bm_pass3_candidates<<<2048, 256, 0, stream>>>(keys, meta, cand, candCount);
    bm_pass4_threshold <<<B_DIM, 256, 0, stream>>>(cand, candCount, meta, thresh);
    bm_pass5_mask      <<<2048, 256, 0, stream>>>(keys, thresh, out);
}